// FNO3D_54099408061102
// MI455X (gfx1250) — compile-verified
//
#include <hip/hip_runtime.h>
#include <math.h>

// ---------------------------------------------------------------------------
// FNO3D forward for MI455X (gfx1250, wave32, WMMA).
// Spectral conv implemented as truncated separable DFTs (64->12 per axis),
// channel-mixing 1x1 convs as fp32 WMMA GEMMs (v_wmma_f32_16x16x4_f32),
// 32 spatial points x 24 output channels per wave, branchless operand setup.
// ---------------------------------------------------------------------------

#define BATCH 4
#define CIN   3
#define WIDTH 24
#define LAYERS 4
#define ZD 64
#define YD 64
#define XD 64
#define SVOL (ZD*YD*XD)          // 262144
#define MODES 12
#define FOURIER_K 8
#define FEAT (CIN + 3 + 6*FOURIER_K)   // 54
#define GN_EPS 1e-5f

typedef float v2f __attribute__((ext_vector_type(2)));
typedef float v8f __attribute__((ext_vector_type(8)));

__device__ __forceinline__ float gelu_exact(float v) {
    return 0.5f * v * (1.0f + erff(v * 0.70710678118654752f));
}

// ---------------------------------------------------------------------------
// Twiddle tables: fwd[n*12+k] = e^{-i 2pi nk/64}, inv[n*12+k] = e^{+i 2pi nk/64}
// ---------------------------------------------------------------------------
__global__ void init_twiddles_kernel(float* __restrict__ fwd, float* __restrict__ inv) {
    int t = blockIdx.x * blockDim.x + threadIdx.x;
    if (t >= 64 * MODES) return;
    int n = t / MODES, k = t % MODES;
    float th = 6.283185307179586f * (float)(n * k) / 64.0f;
    float s, c;
    __sincosf(th, &s, &c);
    fwd[2*t + 0] = c;  fwd[2*t + 1] = -s;
    inv[2*t + 0] = c;  inv[2*t + 1] =  s;
}

// ---------------------------------------------------------------------------
// Lift: per-point Fourier positional encoding (54 feats) + GEMV 54->24
// ---------------------------------------------------------------------------
__global__ void lift_kernel(const float* __restrict__ x,
                            const float* __restrict__ w,   // [24][54]
                            const float* __restrict__ bias, // [24]
                            float* __restrict__ h) {
    int t = blockIdx.x * blockDim.x + threadIdx.x;   // over BATCH*SVOL
    if (t >= BATCH * SVOL) return;
    int b = t / SVOL;
    int s = t - b * SVOL;
    int z = s >> 12, y = (s >> 6) & 63, xc = s & 63;

    float feats[FEAT];
    feats[0] = x[(b*CIN + 0) * SVOL + s];
    feats[1] = x[(b*CIN + 1) * SVOL + s];
    feats[2] = x[(b*CIN + 2) * SVOL + s];
    float fz = -1.0f + 2.0f * (float)z  / 63.0f;
    float fy = -1.0f + 2.0f * (float)y  / 63.0f;
    float fx = -1.0f + 2.0f * (float)xc / 63.0f;
    feats[3] = fz; feats[4] = fy; feats[5] = fx;
    #pragma unroll
    for (int k = 1; k <= FOURIER_K; ++k) {
        float wk = (float)k * 3.14159265358979323846f;
        float sz, cz, sy, cy, sx, cx;
        __sincosf(wk * fz, &sz, &cz);
        __sincosf(wk * fy, &sy, &cy);
        __sincosf(wk * fx, &sx, &cx);
        int o = 6 + (k - 1) * 6;
        feats[o+0] = sz; feats[o+1] = cz;
        feats[o+2] = sy; feats[o+3] = cy;
        feats[o+4] = sx; feats[o+5] = cx;
    }
    #pragma unroll 4
    for (int o = 0; o < WIDTH; ++o) {
        float acc = bias[o];
        const float* wr = w + o * FEAT;
        #pragma unroll
        for (int c = 0; c < FEAT; ++c) acc = fmaf(feats[c], wr[c], acc);
        h[(b * WIDTH + o) * SVOL + s] = acc;
    }
}

// ---------------------------------------------------------------------------
// 1x1 conv as WMMA GEMM: out[b,o,s] = sum_c in[b,c,s]*w[o,c] + bias[o]
// Per wave: 32 spatial points x 24 output channels -> four 16x16 f32 tiles,
// K-loop over 24 input channels in steps of 4 (v_wmma_f32_16x16x4_f32).
// A layout (ISA 7.12.2, 32-bit A 16x4): lanes 0-15 VGPR0=K0,VGPR1=K1;
// lanes 16-31 VGPR0=K2,VGPR1=K3.  B (4x16) mirrors along N.
// Branchless: second N-tile (chans 16..23) uses clamped index + 0/1 mask so
// EXEC never diverges around the WMMA chain.
// ---------------------------------------------------------------------------
__global__ __launch_bounds__(256)
void conv1x1_wmma_kernel(const float* __restrict__ in,
                         const float* __restrict__ w,    // [24][24] (o-major)
                         const float* __restrict__ bias, // [24]
                         float* __restrict__ out,
                         int apply_gelu) {
    const int NS = BATCH * SVOL;
    int wave = blockIdx.x * (blockDim.x >> 5) + (threadIdx.x >> 5);
    int lane = threadIdx.x & 31;
    int p0 = wave * 32;                      // 32 contiguous points per wave
    if (p0 >= NS) return;                    // wave-uniform: EXEC stays all-ones
    int b  = p0 / SVOL;
    int s0 = p0 - b * SVOL;                  // same batch for all 32 (SVOL%32==0)
    const float* inb = in + (size_t)b * WIDTH * SVOL + s0;

    int pt = lane & 15;        // A-matrix M (spatial point within tile)
    int kh = lane >> 4;        // lane-half selects K pair
    int n  = lane & 15;        // B/C/D N index (output channel within tile)
    int n2 = 16 + (n & 7);     // clamped second-tile channel (always in-bounds)
    float msk = (n < 8) ? 1.0f : 0.0f;

    // Prefetch next wave-group's activations (hide strided A-load latency)
    __builtin_prefetch(inb + 32, 0, 3);

    v8f accA0, accA1, accB0, accB1;          // {M-tile A/B} x {N-tile 0/1}
    float b0 = bias[n];
    float b1 = bias[n2] * msk;
    #pragma unroll
    for (int r = 0; r < 8; ++r) {
        accA0[r] = b0; accA1[r] = b1;
        accB0[r] = b0; accB1[r] = b1;
    }

    #pragma unroll
    for (int k0 = 0; k0 < WIDTH; k0 += 4) {
        int ka = k0 + 2 * kh;
        v2f aLo, aHi;
        aLo.x = inb[(ka + 0) * SVOL + pt];
        aLo.y = inb[(ka + 1) * SVOL + pt];
        aHi.x = inb[(ka + 0) * SVOL + 16 + pt];
        aHi.y = inb[(ka + 1) * SVOL + 16 + pt];
        v2f bv0, bv1;
        bv0.x = w[n  * WIDTH + ka + 0];
        bv0.y = w[n  * WIDTH + ka + 1];
        bv1.x = w[n2 * WIDTH + ka + 0] * msk;
        bv1.y = w[n2 * WIDTH + ka + 1] * msk;
        accA0 = __builtin_amdgcn_wmma_f32_16x16x4_f32(false, aLo, false, bv0,
                                                      (short)0, accA0, false, false);
        accA1 = __builtin_amdgcn_wmma_f32_16x16x4_f32(false, aLo, false, bv1,
                                                      (short)0, accA1, false, false);
        accB0 = __builtin_amdgcn_wmma_f32_16x16x4_f32(false, aHi, false, bv0,
                                                      (short)0, accB0, false, false);
        accB1 = __builtin_amdgcn_wmma_f32_16x16x4_f32(false, aHi, false, bv1,
                                                      (short)0, accB1, false, false);
    }

    // D layout: VGPR r: lanes0-15 -> M=r, lanes16-31 -> M=8+r; N = lane&15
    int mbase = kh * 8;
    float* outb = out + (size_t)b * WIDTH * SVOL + s0;
    #pragma unroll
    for (int r = 0; r < 8; ++r) {
        int sp = mbase + r;
        float vA0 = accA0[r], vB0 = accB0[r];
        if (apply_gelu) { vA0 = gelu_exact(vA0); vB0 = gelu_exact(vB0); }
        outb[n * SVOL + sp]      = vA0;
        outb[n * SVOL + 16 + sp] = vB0;
        if (n < 8) {
            float vA1 = accA1[r], vB1 = accB1[r];
            if (apply_gelu) { vA1 = gelu_exact(vA1); vB1 = gelu_exact(vB1); }
            outb[(16 + n) * SVOL + sp]      = vA1;
            outb[(16 + n) * SVOL + 16 + sp] = vB1;
        }
    }
}

// ---------------------------------------------------------------------------
// Depthwise 3x3x3 (SAME, zero pad), accumulate into yacc (+= bias + conv)
// ---------------------------------------------------------------------------
__global__ void depthwise_kernel(const float* __restrict__ h,
                                 const float* __restrict__ w,    // [24][27]
                                 const float* __restrict__ bias, // [24]
                                 float* __restrict__ yacc) {
    int t = blockIdx.x * blockDim.x + threadIdx.x;   // over BATCH*WIDTH*SVOL
    if (t >= BATCH * WIDTH * SVOL) return;
    int s  = t & (SVOL - 1);
    int bc = t >> 18;                  // SVOL = 2^18
    int c  = bc % WIDTH;
    int z = s >> 12, y = (s >> 6) & 63, xc = s & 63;
    const float* hb = h + (size_t)bc * SVOL;
    const float* wc = w + c * 27;
    float acc = bias[c];
    #pragma unroll
    for (int dz = -1; dz <= 1; ++dz) {
        int zz = z + dz;
        if (zz < 0 || zz >= ZD) continue;
        #pragma unroll
        for (int dy = -1; dy <= 1; ++dy) {
            int yy = y + dy;
            if (yy < 0 || yy >= YD) continue;
            #pragma unroll
            for (int dx = -1; dx <= 1; ++dx) {
                int xx = xc + dx;
                if (xx < 0 || xx >= XD) continue;
                acc = fmaf(wc[(dz+1)*9 + (dy+1)*3 + (dx+1)],
                           hb[(zz << 12) + (yy << 6) + xx], acc);
            }
        }
    }
    yacc[t] += acc;
}

// ---------------------------------------------------------------------------
// Forward x-rDFT: real [line][64] -> complex [line][12]
// ---------------------------------------------------------------------------
__global__ void fdft_x_kernel(const float* __restrict__ h,
                              const float* __restrict__ tw,
                              float* __restrict__ o) {
    int t = blockIdx.x * blockDim.x + threadIdx.x;   // lines*12
    const int total = BATCH * WIDTH * ZD * YD * MODES;
    if (t >= total) return;
    int k = t % MODES;
    int line = t / MODES;
    const float* p = h + (size_t)line * 64;
    float re = 0.0f, im = 0.0f;
    #pragma unroll 8
    for (int nn = 0; nn < 64; ++nn) {
        float v = p[nn];
        re = fmaf(v, tw[(nn*MODES + k)*2 + 0], re);
        im = fmaf(v, tw[(nn*MODES + k)*2 + 1], im);
    }
    o[2*t + 0] = re;
    o[2*t + 1] = im;
}

// Forward complex DFT over middle axis: [nouter][64][ninner] -> [nouter][12][ninner]
__global__ void dft_fwd_c_kernel(const float* __restrict__ in,
                                 const float* __restrict__ tw,
                                 float* __restrict__ out,
                                 int nouter, int ninner) {
    int t = blockIdx.x * blockDim.x + threadIdx.x;
    int total = nouter * MODES * ninner;
    if (t >= total) return;
    int inn = t % ninner;
    int r   = t / ninner;
    int k   = r % MODES;
    int ou  = r / MODES;
    const float* p = in + ((size_t)ou * 64 * ninner + inn) * 2;
    float re = 0.0f, im = 0.0f;
    #pragma unroll 4
    for (int nn = 0; nn < 64; ++nn) {
        float ar = p[(size_t)nn * ninner * 2 + 0];
        float ai = p[(size_t)nn * ninner * 2 + 1];
        float tr = tw[(nn*MODES + k)*2 + 0];
        float ti = tw[(nn*MODES + k)*2 + 1];
        re += ar*tr - ai*ti;
        im += ar*ti + ai*tr;
    }
    size_t oidx = ((size_t)ou * MODES + k) * ninner + inn;
    out[2*oidx + 0] = re;
    out[2*oidx + 1] = im;
}

// Inverse complex DFT expansion: [nouter][12][ninner] -> [nouter][64][ninner]
__global__ void dft_inv_c_kernel(const float* __restrict__ in,
                                 const float* __restrict__ tw,
                                 float* __restrict__ out,
                                 int nouter, int ninner) {
    int t = blockIdx.x * blockDim.x + threadIdx.x;
    int total = nouter * 64 * ninner;
    if (t >= total) return;
    int inn = t % ninner;
    int r   = t / ninner;
    int z   = r % 64;
    int ou  = r / 64;
    const float* p = in + ((size_t)ou * MODES * ninner + inn) * 2;
    float re = 0.0f, im = 0.0f;
    #pragma unroll
    for (int k = 0; k < MODES; ++k) {
        float ar = p[(size_t)k * ninner * 2 + 0];
        float ai = p[(size_t)k * ninner * 2 + 1];
        float tr = tw[(z*MODES + k)*2 + 0];
        float ti = tw[(z*MODES + k)*2 + 1];
        re += ar*tr - ai*ti;
        im += ar*ti + ai*tr;
    }
    size_t oidx = ((size_t)ou * 64 + z) * ninner + inn;
    out[2*oidx + 0] = re;
    out[2*oidx + 1] = im;
}

// Per-mode complex channel mixing: out[b,o,m] = sum_c hf[b,c,m] * W[c,o,m]
__global__ void specmul_kernel(const float* __restrict__ hf,
                               const float* __restrict__ w,   // [c][o][1728][2]
                               float* __restrict__ of) {
    const int NM = MODES * MODES * MODES;   // 1728
    int t = blockIdx.x * blockDim.x + threadIdx.x;
    int total = BATCH * WIDTH * NM;
    if (t >= total) return;
    int m = t % NM;
    int r = t / NM;
    int o = r % WIDTH;
    int b = r / WIDTH;
    float re = 0.0f, im = 0.0f;
    #pragma unroll 4
    for (int c = 0; c < WIDTH; ++c) {
        size_t hi = ((size_t)(b*WIDTH + c) * NM + m) * 2;
        size_t wi = ((size_t)(c*WIDTH + o) * NM + m) * 2;
        float ar = hf[hi], ai = hf[hi+1];
        float wr = w[wi],  wi2 = w[wi+1];
        re += ar*wr - ai*wi2;
        im += ar*wi2 + ai*wr;
    }
    size_t oi = ((size_t)(b*WIDTH + o) * NM + m) * 2;
    of[oi]   = re;
    of[oi+1] = im;
}

// Inverse x-rDFT (half-spectrum, 12 modes) fused with accumulate into yacc.
// x_n = (1/64^3) * (Re X0 + sum_{k=1..11} 2*(re_k cos - im_k sin))
__global__ void idft_x_add_kernel(const float* __restrict__ in,   // [line][12] cplx
                                  const float* __restrict__ tw,   // inverse twiddles
                                  float* __restrict__ yacc) {
    int t = blockIdx.x * blockDim.x + threadIdx.x;   // over BATCH*WIDTH*SVOL
    if (t >= BATCH * WIDTH * SVOL) return;
    int xc   = t & 63;
    int line = t >> 6;
    const float* p = in + (size_t)line * MODES * 2;
    float acc = p[0];                                // Re(X0)
    #pragma unroll
    for (int k = 1; k < MODES; ++k) {
        float tr = tw[(xc*MODES + k)*2 + 0];
        float ti = tw[(xc*MODES + k)*2 + 1];
        acc += 2.0f * (p[2*k]*tr - p[2*k+1]*ti);
    }
    yacc[t] += acc * (1.0f / 262144.0f);             // 1/64^3
}

// h = h + gamma * gelu(yacc)
__global__ void residual_kernel(float* __restrict__ h,
                                const float* __restrict__ yacc,
                                const float* __restrict__ gammas, int layer) {
    int t = blockIdx.x * blockDim.x + threadIdx.x;
    if (t >= BATCH * WIDTH * SVOL) return;
    float g = gammas[layer];
    h[t] = fmaf(g, gelu_exact(yacc[t]), h[t]);
}

// ---------------------------------------------------------------------------
// GroupNorm (1 group): deterministic two-level tree reduction
// ---------------------------------------------------------------------------
#define GN_NB 768          // blocks per batch: 24*SVOL / (256*32)
__global__ void gn_partial_kernel(const float* __restrict__ h,
                                  float* __restrict__ part) {
    __shared__ float ss[256], s2[256];
    int b   = blockIdx.y;
    int blk = blockIdx.x;
    size_t base = (size_t)b * WIDTH * SVOL + (size_t)blk * 8192 + threadIdx.x;
    float s = 0.0f, q = 0.0f;
    #pragma unroll 8
    for (int i = 0; i < 32; ++i) {
        float v = h[base + (size_t)i * 256];
        s += v; q = fmaf(v, v, q);
    }
    ss[threadIdx.x] = s; s2[threadIdx.x] = q;
    __syncthreads();
    for (int st = 128; st > 0; st >>= 1) {
        if (threadIdx.x < st) {
            ss[threadIdx.x] += ss[threadIdx.x + st];
            s2[threadIdx.x] += s2[threadIdx.x + st];
        }
        __syncthreads();
    }
    if (threadIdx.x == 0) {
        part[(b * GN_NB + blk) * 2 + 0] = ss[0];
        part[(b * GN_NB + blk) * 2 + 1] = s2[0];
    }
}

__global__ void gn_finalize_kernel(const float* __restrict__ part,
                                   float* __restrict__ stats) {
    __shared__ float ss[256], s2[256];
    int b = blockIdx.x;
    float s = 0.0f, q = 0.0f;
    for (int i = threadIdx.x; i < GN_NB; i += 256) {
        s += part[(b * GN_NB + i) * 2 + 0];
        q += part[(b * GN_NB + i) * 2 + 1];
    }
    ss[threadIdx.x] = s; s2[threadIdx.x] = q;
    __syncthreads();
    for (int st = 128; st > 0; st >>= 1) {
        if (threadIdx.x < st) {
            ss[threadIdx.x] += ss[threadIdx.x + st];
            s2[threadIdx.x] += s2[threadIdx.x + st];
        }
        __syncthreads();
    }
    if (threadIdx.x == 0) {
        const float invN = 1.0f / (float)(WIDTH * SVOL);
        float m = ss[0] * invN;
        float v = s2[0] * invN - m * m;
        stats[b * 2 + 0] = m;
        stats[b * 2 + 1] = rsqrtf(v + GN_EPS);
    }
}

__global__ void gn_apply_kernel(float* __restrict__ h,
                                const float* __restrict__ stats,
                                const float* __restrict__ gw,
                                const float* __restrict__ gb) {
    int t = blockIdx.x * blockDim.x + threadIdx.x;
    if (t >= BATCH * WIDTH * SVOL) return;
    int bc = t >> 18;
    int c  = bc % WIDTH;
    int b  = bc / WIDTH;
    float m = stats[b*2 + 0], r = stats[b*2 + 1];
    h[t] = fmaf((h[t] - m) * r, gw[c], gb[c]);
}

// Final projection 24 -> 1
__global__ void proj2_kernel(const float* __restrict__ p,
                             const float* __restrict__ w,   // [1][24]
                             const float* __restrict__ bias,
                             float* __restrict__ out) {
    int t = blockIdx.x * blockDim.x + threadIdx.x;   // over BATCH*SVOL
    if (t >= BATCH * SVOL) return;
    int b = t / SVOL;
    int s = t - b * SVOL;
    float acc = bias[0];
    #pragma unroll
    for (int c = 0; c < WIDTH; ++c)
        acc = fmaf(p[(size_t)(b*WIDTH + c) * SVOL + s], w[c], acc);
    out[t] = acc;
}

// ---------------------------------------------------------------------------
// Host launcher
// ---------------------------------------------------------------------------
extern "C" void kernel_launch(void* const* d_in, const int* in_sizes, int n_in,
                              void* d_out, int out_size, void* d_ws, size_t ws_size,
                              hipStream_t stream) {
    const float* x       = (const float*)d_in[0];
    const float* lift_w  = (const float*)d_in[1];
    const float* lift_b  = (const float*)d_in[2];
    const float* spec_w  = (const float*)d_in[3];
    const float* ws_w    = (const float*)d_in[4];
    const float* ws_b    = (const float*)d_in[5];
    const float* loc_w   = (const float*)d_in[6];
    const float* loc_b   = (const float*)d_in[7];
    const float* gn_w    = (const float*)d_in[8];
    const float* gn_b    = (const float*)d_in[9];
    const float* gammas  = (const float*)d_in[10];
    const float* proj1_w = (const float*)d_in[11];
    const float* proj1_b = (const float*)d_in[12];
    const float* proj2_w = (const float*)d_in[13];
    const float* proj2_b = (const float*)d_in[14];
    float* out = (float*)d_out;

    // Workspace layout (bytes)
    const size_t HB  = (size_t)BATCH * WIDTH * SVOL * sizeof(float);  // 100.7 MB
    const size_t S1B = (size_t)BATCH * WIDTH * ZD * YD * MODES * 2 * sizeof(float); // 37.7 MB
    const size_t S2B = (size_t)BATCH * WIDTH * ZD * MODES * MODES * 2 * sizeof(float); // 7.1 MB
    const size_t S3B = (size_t)BATCH * WIDTH * MODES * MODES * MODES * 2 * sizeof(float); // 1.3 MB
    char* ws = (char*)d_ws;
    size_t off = 0;
    float* h    = (float*)(ws + off); off += HB;
    float* yacc = (float*)(ws + off); off += HB;
    float* S1   = (float*)(ws + off); off += S1B;   // also reused as S6
    float* S2   = (float*)(ws + off); off += S2B;   // also reused as S5
    float* S3   = (float*)(ws + off); off += S3B;
    float* S4   = (float*)(ws + off); off += S3B;
    float* twf  = (float*)(ws + off); off += 64 * MODES * 2 * sizeof(float);
    float* twi  = (float*)(ws + off); off += 64 * MODES * 2 * sizeof(float);
    float* part = (float*)(ws + off); off += BATCH * GN_NB * 2 * sizeof(float);
    float* stats= (float*)(ws + off); off += BATCH * 2 * sizeof(float);

    const int T = 256;
    const int NBS   = (BATCH * SVOL) / T;                 // 4096
    const int NFULL = (BATCH * WIDTH * SVOL) / T;         // 98304
    const int NWAVE = (BATCH * SVOL / 32) / (T / 32);     // 4096 blocks, 32 pts/wave

    init_twiddles_kernel<<<3, T, 0, stream>>>(twf, twi);
    lift_kernel<<<NBS, T, 0, stream>>>(x, lift_w, lift_b, h);

    for (int i = 0; i < LAYERS; ++i) {
        // yacc = conv1x1(h, ws_w[i]) + ws_b[i]        (WMMA)
        conv1x1_wmma_kernel<<<NWAVE, T, 0, stream>>>(
            h, ws_w + i * WIDTH * WIDTH, ws_b + i * WIDTH, yacc, 0);
        // yacc += depthwise3(h) + loc_b[i]
        depthwise_kernel<<<NFULL, T, 0, stream>>>(
            h, loc_w + i * WIDTH * 27, loc_b + i * WIDTH, yacc);

        // Spectral path: truncated separable DFTs
        fdft_x_kernel<<<(BATCH*WIDTH*ZD*YD*MODES)/T, T, 0, stream>>>(h, twf, S1);
        dft_fwd_c_kernel<<<(BATCH*WIDTH*ZD*MODES*MODES)/T, T, 0, stream>>>(
            S1, twf, S2, BATCH*WIDTH*ZD, MODES);                 // over y
        dft_fwd_c_kernel<<<(BATCH*WIDTH*MODES*MODES*MODES)/T, T, 0, stream>>>(
            S2, twf, S3, BATCH*WIDTH, MODES*MODES);              // over z
        specmul_kernel<<<(BATCH*WIDTH*MODES*MODES*MODES)/T, T, 0, stream>>>(
            S3, spec_w + (size_t)i * WIDTH * WIDTH * MODES * MODES * MODES * 2, S4);
        dft_inv_c_kernel<<<(BATCH*WIDTH*ZD*MODES*MODES)/T, T, 0, stream>>>(
            S4, twi, S2, BATCH*WIDTH, MODES*MODES);              // expand z
        dft_inv_c_kernel<<<(BATCH*WIDTH*ZD*YD*MODES)/T, T, 0, stream>>>(
            S2, twi, S1, BATCH*WIDTH*ZD, MODES);                 // expand y
        idft_x_add_kernel<<<NFULL, T, 0, stream>>>(S1, twi, yacc);

        // h = h + gamma * gelu(yacc); GroupNorm(1 group)
        residual_kernel<<<NFULL, T, 0, stream>>>(h, yacc, gammas, i);
        gn_partial_kernel<<<dim3(GN_NB, BATCH), T, 0, stream>>>(h, part);
        gn_finalize_kernel<<<BATCH, T, 0, stream>>>(part, stats);
        gn_apply_kernel<<<NFULL, T, 0, stream>>>(h, stats, gn_w + i*WIDTH, gn_b + i*WIDTH);
    }

    // p = gelu(conv1x1(h, proj1)) (WMMA, fused GELU) ; out = conv1x1(p, proj2)
    conv1x1_wmma_kernel<<<NWAVE, T, 0, stream>>>(h, proj1_w, proj1_b, yacc, 1);
    proj2_kernel<<<NBS, T, 0, stream>>>(yacc, proj2_w, proj2_b, out);
}